// GCN_50328426775032
// MI455X (gfx1250) — compile-verified
//
#include <hip/hip_runtime.h>

typedef __attribute__((ext_vector_type(2))) float v2f;
typedef __attribute__((ext_vector_type(8))) float v8f;

#define DIM 128

// ---- Stage 1: degree (self-loop weight 1 + scatter-add of edge weights) ----
__global__ __launch_bounds__(256) void k_deg_init(float* __restrict__ deg, int n) {
  int i = blockIdx.x * blockDim.x + threadIdx.x;
  if (i < n) deg[i] = 1.0f;
}

__global__ __launch_bounds__(256) void k_deg_scatter(const int* __restrict__ src,
                                                     const float* __restrict__ w,
                                                     float* __restrict__ deg, int E) {
  int e = blockIdx.x * blockDim.x + threadIdx.x;
  if (e < E) atomicAdd(&deg[src[e]], w[e]);
}

__global__ __launch_bounds__(256) void k_dinv(const float* __restrict__ deg,
                                              float* __restrict__ dinv, int n) {
  int i = blockIdx.x * blockDim.x + threadIdx.x;
  if (i < n) dinv[i] = rsqrtf(deg[i]);
}

// ---- Stage 2: Y = X @ W^T via V_WMMA_F32_16X16X4_F32 (full f32 precision) ----
// Block = 256 threads = 8 waves. Block handles a 16-row strip of X;
// wave w computes the 16x16 output tile for columns [16w, 16w+16).
__global__ __launch_bounds__(256) void k_gemm_xwT(const float* __restrict__ X,
                                                  const float* __restrict__ W,
                                                  float* __restrict__ Y, int n) {
  const int wave = threadIdx.x >> 5;
  const int lane = threadIdx.x & 31;
  const int row0 = blockIdx.x << 4;
  const int m    = lane & 15;        // A row within tile / C column
  const int hi   = lane >> 4;        // lane half
  const int kb   = hi << 1;          // K sub-offset: 0 or 2
  const int col  = (wave << 4) + m;  // B/C column (output feature)

  int row = row0 + m;
  if (row >= n) row = n - 1;  // clamp (harmless duplicate reads; EXEC stays all-1s for WMMA)
  const float* arow = X + (size_t)row * DIM;
  const float* brow = W + (size_t)col * DIM;  // B[k][col] = W[col][k]

  v8f c = {};
#pragma unroll
  for (int k0 = 0; k0 < DIM; k0 += 4) {
    v2f a = *(const v2f*)(arow + k0 + kb);  // A[m][k0+kb], A[m][k0+kb+1]
    v2f b = *(const v2f*)(brow + k0 + kb);  // B[k0+kb][col], B[k0+kb+1][col]
    c = __builtin_amdgcn_wmma_f32_16x16x4_f32(false, a, false, b, (short)0, c,
                                              false, false);
  }

  const int mh = hi << 3;  // C/D: M = v + 8*(lane>>4)
#pragma unroll
  for (int v = 0; v < 8; ++v) {
    int r = row0 + mh + v;
    if (r < n) Y[(size_t)r * DIM + col] = c[v];
  }
}

// ---- Stage 3: out = dinv[i]^2 * Y[i] + b  (self-loop contribution + bias) ----
__global__ __launch_bounds__(256) void k_selfloop_bias(const float* __restrict__ Y,
                                                       const float* __restrict__ dinv,
                                                       const float* __restrict__ bias,
                                                       float* __restrict__ out, int total) {
  int idx = blockIdx.x * blockDim.x + threadIdx.x;
  if (idx >= total) return;
  int i = idx >> 7;          // node
  int d = idx & (DIM - 1);   // feature
  float di = dinv[i];
  out[idx] = Y[idx] * di * di + bias[d];
}

// ---- Stage 4: edge gather-scale-scatter: out[tgt] += wn * Y[src] ----
// One wave per edge: 32 lanes x float4 = 128 dims; 4 atomicAdd f32 per lane.
// Y and out both resident in 192MB L2 -> runs at L2/atomic throughput.
__global__ __launch_bounds__(256) void k_edge_scatter(const int* __restrict__ src,
                                                      const int* __restrict__ tgt,
                                                      const float* __restrict__ w,
                                                      const float* __restrict__ dinv,
                                                      const float* __restrict__ Y,
                                                      float* __restrict__ out, int E) {
  int e = blockIdx.x * 8 + (threadIdx.x >> 5);
  if (e >= E) return;
  int lane = threadIdx.x & 31;
  int s = src[e];
  int t = tgt[e];
  float wn = w[e] * dinv[s] * dinv[t];
  const float4 y = *(const float4*)(Y + (size_t)s * DIM + (lane << 2));
  float* op = out + (size_t)t * DIM + (lane << 2);
  atomicAdd(op + 0, y.x * wn);
  atomicAdd(op + 1, y.y * wn);
  atomicAdd(op + 2, y.z * wn);
  atomicAdd(op + 3, y.w * wn);
}

extern "C" void kernel_launch(void* const* d_in, const int* in_sizes, int n_in,
                              void* d_out, int out_size, void* d_ws, size_t ws_size,
                              hipStream_t stream) {
  const float* x          = (const float*)d_in[0];
  const int*   edge_index = (const int*)d_in[1];
  const float* ew         = (const float*)d_in[2];
  const float* W          = (const float*)d_in[3];
  const float* b          = (const float*)d_in[4];

  const int N = in_sizes[0] / DIM;
  const int E = in_sizes[2];
  const int* src = edge_index;      // edge_index[0, :]
  const int* tgt = edge_index + E;  // edge_index[1, :]

  float* ws   = (float*)d_ws;
  float* deg  = ws;                       // N floats
  float* dinv = ws + N;                   // N floats
  float* Y    = ws + 2 * (size_t)N;       // N*128 floats (X @ W^T)
  float* out  = (float*)d_out;

  k_deg_init<<<(N + 255) / 256, 256, 0, stream>>>(deg, N);
  k_deg_scatter<<<(E + 255) / 256, 256, 0, stream>>>(src, ew, deg, E);
  k_dinv<<<(N + 255) / 256, 256, 0, stream>>>(deg, dinv, N);
  k_gemm_xwT<<<(N + 15) / 16, 256, 0, stream>>>(x, W, Y, N);
  int total = N * DIM;
  k_selfloop_bias<<<(total + 255) / 256, 256, 0, stream>>>(Y, dinv, b, out, total);
  k_edge_scatter<<<(E + 7) / 8, 256, 0, stream>>>(src, tgt, ew, dinv, Y, out, E);
}